// L1Conv2D_19009525252352
// MI455X (gfx1250) — compile-verified
//
#include <hip/hip_runtime.h>
#include <math.h>

#define BATCH 4
#define H     32
#define W     32
#define CIN   64
#define COUT  64
#define KDIM  3
#define TILE_COLS 34                       // cols -1..32 stored at col+1
#define LDS_FLOATS (3 * TILE_COLS * CIN)   // 6528 floats = 26112 B

// ---- gfx1250 async global->LDS path (guarded; falls back to sync copy) ----
#if defined(__has_builtin)
#if __has_builtin(__builtin_amdgcn_global_load_async_to_lds_b128) && \
    __has_builtin(__builtin_amdgcn_s_wait_asynccnt)
#define HAVE_ASYNC_LDS 1
#endif
#endif

#ifdef HAVE_ASYNC_LDS
// Builtin wants pointers to 16-byte int vectors (per hipcc diagnostic).
typedef int v4i __attribute__((vector_size(16)));
typedef __attribute__((address_space(1))) v4i* gv4i_p;   // global
typedef __attribute__((address_space(3))) v4i* lv4i_p;   // LDS
#endif

__global__ __launch_bounds__(256) void l1conv_kernel(
    const float* __restrict__ x,      // [B,H,W,CIN]
    const float* __restrict__ wgt,    // [KDIM,KDIM,CIN,COUT]
    float* __restrict__ out)          // [B,H,W,COUT]
{
    __shared__ float xs[LDS_FLOATS];

    const int tid = threadIdx.x;
    const int bh  = blockIdx.x;       // 0..BATCH*H-1
    const int b   = bh >> 5;          // /H (H==32)
    const int h   = bh & 31;

    // Prefetch weight tensor into cache (gfx1250 global_prefetch_b8).
    for (int off = tid * 32; off < KDIM * KDIM * CIN * COUT; off += 256 * 32)
        __builtin_prefetch(wgt + off, 0, 0);

    // ---- Stage zero-padded x tile: rows h-1..h+1, cols -1..32, all 64 ch ----
    const int totalUnits = 3 * TILE_COLS * (CIN / 4);   // float4 units = 1632
    for (int u = tid; u < totalUnits; u += 256) {
        const int c4  = u & 15;          // CIN/4 == 16
        const int rc  = u >> 4;
        const int col = rc % TILE_COLS;  // 0..33  (input col = col-1)
        const int r3  = rc / TILE_COLS;  // 0..2   (input row = h-1+r3)
        const int gr  = h - 1 + r3;
        const int gc  = col - 1;
        const int ldsOff = (r3 * TILE_COLS + col) * CIN + c4 * 4;
        if ((unsigned)gr < (unsigned)H && (unsigned)gc < (unsigned)W) {
            const float* gp = x + (((b * H + gr) * W + gc) * CIN + c4 * 4);
#ifdef HAVE_ASYNC_LDS
            __builtin_amdgcn_global_load_async_to_lds_b128(
                (gv4i_p)(float*)gp, (lv4i_p)&xs[ldsOff], 0, 0);
#else
            *(float4*)&xs[ldsOff] = *(const float4*)gp;
#endif
        } else {
            *(float4*)&xs[ldsOff] = make_float4(0.f, 0.f, 0.f, 0.f);
        }
    }
#ifdef HAVE_ASYNC_LDS
    __builtin_amdgcn_s_wait_asynccnt(0);
#endif
    __syncthreads();

    // ---- Compute: wave owns 4 pixels, lane owns 2 couts ----
    const int lane = tid & 31;
    const int wave = tid >> 5;
    const int p0   = wave * 4;        // first output col of this wave
    const int co   = lane * 2;        // cout pair

    float acc[4][2] = {};

    for (int ki = 0; ki < KDIM; ++ki) {
        for (int kj = 0; kj < KDIM; ++kj) {
            const float* wk = wgt + ((ki * KDIM + kj) * CIN) * COUT + co;
            const int xbase = (ki * TILE_COLS) * CIN;
#pragma unroll 4
            for (int c = 0; c < CIN; c += 4) {
                // coalesced b64 weight loads (lane l -> bytes 8l..8l+7)
                const float2 w0 = *(const float2*)(wk + (c + 0) * COUT);
                const float2 w1 = *(const float2*)(wk + (c + 1) * COUT);
                const float2 w2 = *(const float2*)(wk + (c + 2) * COUT);
                const float2 w3 = *(const float2*)(wk + (c + 3) * COUT);
#pragma unroll
                for (int p = 0; p < 4; ++p) {
                    // wave-uniform LDS broadcast, 4 channels at once
                    const float4 xv =
                        *(const float4*)&xs[xbase + (p0 + p + kj) * CIN + c];
                    acc[p][0] += fabsf(xv.x - w0.x) + fabsf(xv.y - w1.x) +
                                 fabsf(xv.z - w2.x) + fabsf(xv.w - w3.x);
                    acc[p][1] += fabsf(xv.x - w0.y) + fabsf(xv.y - w1.y) +
                                 fabsf(xv.z - w2.y) + fabsf(xv.w - w3.y);
                }
            }
        }
    }

    // ---- Coalesced b64 stores ----
    float* op = out + ((b * H + h) * W + p0) * COUT + co;
#pragma unroll
    for (int p = 0; p < 4; ++p)
        *(float2*)(op + p * COUT) = make_float2(acc[p][0], acc[p][1]);
}

extern "C" void kernel_launch(void* const* d_in, const int* in_sizes, int n_in,
                              void* d_out, int out_size, void* d_ws, size_t ws_size,
                              hipStream_t stream)
{
    const float* x   = (const float*)d_in[0];   // (4,32,32,64) f32
    const float* wgt = (const float*)d_in[1];   // (3,3,64,64)  f32
    float*       out = (float*)d_out;           // (4,32,32,64) f32
    (void)in_sizes; (void)n_in; (void)out_size; (void)d_ws; (void)ws_size;

    dim3 grid(BATCH * H);   // 128 blocks, one per (b, h) row
    dim3 block(256);        // 8 waves (wave32)
    l1conv_kernel<<<grid, block, 0, stream>>>(x, wgt, out);
}